// TriplaneEncoder_70901320122780
// MI455X (gfx1250) — compile-verified
//
#include <hip/hip_runtime.h>
#include <stdint.h>

// ---------------------------------------------------------------------------
// Triplane encoder: out[n, c] = bilerp(T_xy[c], (x,y)) + bilerp(T_yz[c], (y,z))
//                             + bilerp(T_zx[c], (z,x))
// Planes are (C=32, H=512, W=512) f32 -> 96 MB total: fully L2-resident
// (192 MB L2), so the bottleneck is scattered L2 gather REQUEST throughput.
// Strategy:
//   * lane = channel -> coalesced 128 B/wave NT output stores (don't evict
//     the L2-resident planes with the 256 MB output stream)
//   * x0/x1 corners are adjacent floats -> one unaligned float2 load per row
//     pair: 6 scattered requests per point per lane instead of 12
//   * one point per wave -> all index/weight math is wave-uniform: pin it to
//     SGPRs via v_readfirstlane so gathers are SADDR + lane-offset VGPR
//   * coords double-buffered into LDS with the CDNA5 async global->LDS engine
//     (global_load_async_to_lds_b128 / s_wait_asynccnt), overlapping the next
//     chunk's coord stream with this chunk's gather math
// ---------------------------------------------------------------------------

#define RESOLUTION 512
#define PLANE_HW   (RESOLUTION * RESOLUTION)
#define NCHAN      32
#define THREADS    256                    // 8 waves (wave32)
#define CHUNK_PTS  128                    // points per staged chunk (1536 B)
#define CHUNKS     4                      // chunks per block
#define BLOCK_PTS  (CHUNK_PTS * CHUNKS)   // 512 points per block
#define PTS_PER_WAVE (CHUNK_PTS / 8)      // 16 points per wave per chunk

typedef __attribute__((address_space(3))) float lds_f32;

// Unaligned-safe 8-byte load (x0,x1 are adjacent columns; parity arbitrary).
__device__ __forceinline__ float2 load2u(const float* p) {
  float2 r;
  __builtin_memcpy(&r, p, sizeof(float2));
  return r;
}

// Wave-uniform-by-construction values -> SGPR (scalarize downstream math).
__device__ __forceinline__ float uniform_f(float v) {
  return __int_as_float(__builtin_amdgcn_readfirstlane(__float_as_int(v)));
}
__device__ __forceinline__ int uniform_i(int v) {
  return __builtin_amdgcn_readfirstlane(v);
}

__device__ __forceinline__ float sample_plane(const float* __restrict__ plane,
                                              unsigned laneOff,
                                              float cx, float cy) {
  const int R = RESOLUTION;
  // torch grid_sample, align_corners=False: fx = ((cx+1)*W - 1) * 0.5
  float fx = ((cx + 1.0f) * (float)R - 1.0f) * 0.5f;
  float fy = ((cy + 1.0f) * (float)R - 1.0f) * 0.5f;
  float x0f = floorf(fx);
  float y0f = floorf(fy);
  float wx1 = fx - x0f;
  float wy1 = fy - y0f;
  float wx0 = 1.0f - wx1;
  float wy0 = 1.0f - wy1;
  int x0 = (int)x0f, y0 = (int)y0f;
  int x1 = x0 + 1,   y1 = y0 + 1;
  // zeros padding: mask weights of out-of-range corners
  if (x0 < 0 || x0 >= R) wx0 = 0.0f;
  if (x1 < 0 || x1 >= R) wx1 = 0.0f;
  if (y0 < 0 || y0 >= R) wy0 = 0.0f;
  if (y1 < 0 || y1 >= R) wy1 = 0.0f;
  // Column PAIR at xcL contains both logical x-corners.
  int xcL = min(max(x0, 0), R - 2);
  int yc0 = min(max(y0, 0), R - 1);
  int yc1 = min(max(y1, 0), R - 1);
  // Row offsets are wave-uniform (one point per wave): pin to SGPRs so each
  // gather address is SADDR(plane) + one VGPR add on the lane offset.
  int off0 = uniform_i(yc0 * R + xcL);
  int off1 = uniform_i(yc1 * R + xcL);
  float2 r0 = load2u(plane + laneOff + off0);   // row y0: cols xcL, xcL+1
  float2 r1 = load2u(plane + laneOff + off1);   // row y1: cols xcL, xcL+1
  // Remap pair elements to logical corners at the clamped boundaries.
  // xc0 == xcL+1 only when x0 >= R-1 ; xc1 == xcL only when x0 < 0.
  bool hi0 = (x0 >= R - 1);
  bool lo1 = (x0 < 0);
  float v00 = hi0 ? r0.y : r0.x;
  float v01 = lo1 ? r0.x : r0.y;
  float v10 = hi0 ? r1.y : r1.x;
  float v11 = lo1 ? r1.x : r1.y;
  return wy0 * (wx0 * v00 + wx1 * v01) + wy1 * (wx0 * v10 + wx1 * v11);
}

__global__ __launch_bounds__(THREADS)
void triplane_encoder_kernel(const float* __restrict__ xyz,
                             const float* __restrict__ Txy,
                             const float* __restrict__ Tyz,
                             const float* __restrict__ Tzx,
                             float* __restrict__ out,
                             int nPoints) {
  __shared__ float s_xyz[2][CHUNK_PTS * 3];

  const int t         = threadIdx.x;
  const int wv        = t >> 5;
  const int lane      = t & 31;
  const int blockBase = blockIdx.x * BLOCK_PTS;

  // Async-stage chunk `c` of this block's coords into LDS buffer `b`.
  // 128 pts * 12 B = 1536 B = 96 x b128 -> lanes 0..95 (waves 0..2, full EXEC).
  auto stage = [&](int c, int b) {
    if (t < (CHUNK_PTS * 3) / 4) {
      int floatOff = t * 4;                                   // within chunk
      int gFloat   = (blockBase + c * CHUNK_PTS) * 3 + floatOff;
      if (gFloat + 4 <= nPoints * 3) {
        lds_f32* lp = (lds_f32*)&s_xyz[b][floatOff];
        unsigned gOffBytes = (unsigned)(gFloat * 4);
        // GVS mode: mem = SGPR64 + VGPR32 ; dest VGPR = LDS byte address
        asm volatile("global_load_async_to_lds_b128 %0, %1, %2"
                     :: "v"(lp), "v"(gOffBytes), "s"(xyz)
                     : "memory");
      }
    }
  };

  stage(0, 0);

  const unsigned laneOff = (unsigned)lane * PLANE_HW;   // channel offset

  for (int c = 0; c < CHUNKS; ++c) {
    // Wait for this chunk's coords (issuing waves drain ASYNCcnt; others
    // pass immediately), then make the LDS buffer visible block-wide.
    asm volatile("s_wait_asynccnt 0x0" ::: "memory");
    __syncthreads();

    // Kick off the next chunk's coord stream while we gather this one.
    if (c + 1 < CHUNKS) stage(c + 1, (c + 1) & 1);

    const float* sx = s_xyz[c & 1];
    const int chunkBase = blockBase + c * CHUNK_PTS;

    #pragma unroll 4
    for (int i = 0; i < PTS_PER_WAVE; ++i) {
      int pl = wv * PTS_PER_WAVE + i;
      int p  = chunkBase + pl;             // wave-uniform
      if (p >= nPoints) break;
      // All lanes read identical LDS addresses -> values are wave-uniform;
      // make that explicit so per-point math lands in SGPRs/scalar form.
      float px = uniform_f(sx[pl * 3 + 0]);
      float py = uniform_f(sx[pl * 3 + 1]);
      float pz = uniform_f(sx[pl * 3 + 2]);
      float acc = sample_plane(Txy, laneOff, px, py)
                + sample_plane(Tyz, laneOff, py, pz)
                + sample_plane(Tzx, laneOff, pz, px);
      // NT store: the 256 MB output stream must not evict the L2-resident
      // planes (the whole point of the 192 MB L2 here).
      __builtin_nontemporal_store(acc, out + (size_t)p * NCHAN + lane);
    }
    // No trailing barrier needed: the next iteration's wait+barrier orders
    // all reads of buffer (c&1) before it is re-staged (chunk c+2) in iteration c+1.
  }
}

extern "C" void kernel_launch(void* const* d_in, const int* in_sizes, int n_in,
                              void* d_out, int out_size, void* d_ws, size_t ws_size,
                              hipStream_t stream) {
  const float* xyz = (const float*)d_in[0];
  const float* Txy = (const float*)d_in[1];
  const float* Tyz = (const float*)d_in[2];
  const float* Tzx = (const float*)d_in[3];
  float* out = (float*)d_out;

  int nPoints = in_sizes[0] / 3;
  int nBlocks = (nPoints + BLOCK_PTS - 1) / BLOCK_PTS;

  triplane_encoder_kernel<<<nBlocks, THREADS, 0, stream>>>(
      xyz, Txy, Tyz, Tzx, out, nPoints);
}